// FNOBlockFramework_84164179132612
// MI455X (gfx1250) — compile-verified
//
#include <hip/hip_runtime.h>
#include <math.h>

typedef __attribute__((ext_vector_type(2))) float v2f;
typedef __attribute__((ext_vector_type(8))) float v8f;

#define FNO_TWO_PI 6.283185307179586f

// ---------------------------------------------------------------------------
// Twiddle / synthesis tables (all transforms have period 256; reduce phase
// mod 256 so sinf/cosf args stay tiny & exact).
//   Fw  [w(256)][32] : cols 0..15 = cos(2*pi*ky*w/256), cols 16..31 = -sin(...)
//   T   [kx(32)][h(256)] : forward along H, kx in {0..15, 240..255}; Tre=cos, Tim=-sin
//   iT  [h(256)][kx(32)] : inverse along H, scaled by 1/(H*W)=1/65536; iTre=cos, iTim=+sin
//   G   [k(32)][w(256)]  : real irfft synthesis: rows 0..15 (Re part): (ky==0?1:2)*cos,
//                          rows 16..31 (Im part): (ky==0?0:-2)*sin   (bin-0 imag dropped)
// ---------------------------------------------------------------------------
__global__ void fno_fill_tables(float* __restrict__ Fw, float* __restrict__ Tre,
                                float* __restrict__ Tim, float* __restrict__ iTre,
                                float* __restrict__ iTim, float* __restrict__ G) {
  int i = blockIdx.x * blockDim.x + threadIdx.x;
  if (i < 8192) {
    int w = i >> 5, c = i & 31, ky = c & 15;
    int p = (ky * w) & 255;
    float th = FNO_TWO_PI * (float)p * (1.0f / 256.0f);
    Fw[i] = (c < 16) ? cosf(th) : -sinf(th);
  } else if (i < 16384) {
    int j = i - 8192;
    int kx = j >> 8, h = j & 255;
    int f = (kx < 16) ? kx : (224 + kx);   // 240 + (kx-16)
    int p = (f * h) & 255;
    float th = FNO_TWO_PI * (float)p * (1.0f / 256.0f);
    Tre[j] = cosf(th);
    Tim[j] = -sinf(th);
  } else if (i < 24576) {
    int j = i - 16384;
    int h = j >> 5, kx = j & 31;
    int f = (kx < 16) ? kx : (224 + kx);
    int p = (f * h) & 255;
    float th = FNO_TWO_PI * (float)p * (1.0f / 256.0f);
    const float s = 1.0f / 65536.0f;       // irfft2 'backward' norm 1/(H*W)
    iTre[j] = cosf(th) * s;
    iTim[j] = sinf(th) * s;
  } else if (i < 32768) {
    int j = i - 24576;
    int k = j >> 8, w = j & 255, ky = k & 15;
    int p = (ky * w) & 255;
    float th = FNO_TWO_PI * (float)p * (1.0f / 256.0f);
    G[j] = (k < 16) ? ((ky == 0 ? 1.0f : 2.0f) * cosf(th))
                    : ((ky == 0 ? 0.0f : -2.0f) * sinf(th));
  }
}

// ---------------------------------------------------------------------------
// Generic strided batched GEMM on v_wmma_f32_16x16x4_f32.
//   C[b] (+)= alpha * A[b] @ B[b]  (+ bias[n]) (-> gelu) (+ res)
// Fragment lane layouts per CDNA5 ISA (wave32):
//   A 16x4 : lane L -> M = L%16, K pair base = 2*(L/16)    (a.x=K, a.y=K+1)
//   B 4x16 : lane L -> N = L%16, K pair base = 2*(L/16)    (b.x=K, b.y=K+1)
//   C/D    : lane L, vgpr v -> M = v + 8*(L/16), N = L%16
// ACONT/BCONT: K-stride==1 -> K pair is one aligned 8-byte load.
// SCALE: apply alpha / M-guard scaling to A (only when alpha!=1 or M%16!=0).
// N2: each wave computes a 16x32 C tile (2 N-subtiles) reusing the A fragment;
//     4 independent accumulator chains (2 N x even/odd K) for MMA latency hiding.
// ---------------------------------------------------------------------------
template <bool ACONT, bool BCONT, bool SCALE, bool N2>
__global__ __launch_bounds__(256) void fno_wmma_gemm(
    const float* __restrict__ A, const float* __restrict__ B, float* __restrict__ C,
    const float* __restrict__ bias, const float* __restrict__ res,
    int M, int N, int K,
    long long sAm, long long sAk, long long bsA,
    long long sBk, long long sBn, long long bsB,
    long long sCm, long long sCn, long long bsC,
    int batch, float alpha, int beta, int act, int tilesM, int tilesN)
{
  int wid  = blockIdx.x * (blockDim.x >> 5) + (threadIdx.x >> 5);
  int nw   = gridDim.x * (blockDim.x >> 5);
  int lane = threadIdx.x & 31;
  int lm   = lane & 15;   // 0..15
  int lh   = lane >> 4;   // 0 or 1
  int tilesNe = N2 ? (tilesN >> 1) : tilesN;
  int tilesMN = tilesM * tilesNe;
  int total   = batch * tilesMN;

  const long long a4 = 4 * sAk;     // A pointer advance per k-step of 4
  const long long b4 = 4 * sBk;
  const long long bn16 = 16 * sBn;  // N-subtile 1 offset in B

  for (int t = wid; t < total; t += nw) {
    int tb = t / tilesMN;
    int r  = t - tb * tilesMN;
    int tm = r / tilesNe;
    int tne = r - tm * tilesNe;
    int tn = N2 ? (tne << 1) : tne;
    const float* Ab = A + (long long)tb * bsA;
    const float* Bb = B + (long long)tb * bsB;
    float*       Cb = C + (long long)tb * bsC;

    int m0 = tm << 4, n0 = tn << 4;
    int cn     = n0 + lm;        // first N-subtile column
    int cmBase = m0 + lh * 8;    // D-layout row base; rows cmBase+0..7

    v8f z = {0.f, 0.f, 0.f, 0.f, 0.f, 0.f, 0.f, 0.f};
    v8f acc00 = z, acc01 = z;    // N-subtile 0: even / odd K chains
    v8f acc10 = z, acc11 = z;    // N-subtile 1
    if (beta) {
      const float* Cp = Cb + (long long)cmBase * sCm + (long long)cn * sCn;
#pragma unroll
      for (int v = 0; v < 8; ++v) {
        bool ok = (cmBase + v < M);
        acc00[v] = ok ? Cp[0] : 0.0f;
        if (N2) acc10[v] = ok ? Cp[16 * sCn] : 0.0f;
        Cp += sCm;
      }
    }

    int am = m0 + lm;
    float as = (am < M) ? alpha : 0.0f;            // fold row-guard into scale
    int amc  = (!SCALE || am < M) ? am : 0;        // clamped in-bounds row
    const float* Ap  = Ab + (long long)amc * sAm + (long long)(lh << 1) * sAk;
    const float* Bp0 = Bb + (long long)cn  * sBn + (long long)(lh << 1) * sBk;
    const float* Bp1 = Bp0 + bn16;

#pragma unroll 2
    for (int k0 = 0; k0 < K; k0 += 8) {
      v2f a0, a1, b00, b01, b10, b11;
      if (ACONT) { a0 = *(const v2f*)Ap; a1 = *(const v2f*)(Ap + a4); }
      else       { a0.x = Ap[0]; a0.y = Ap[sAk]; a1.x = Ap[a4]; a1.y = Ap[a4 + sAk]; }
      if (BCONT) { b00 = *(const v2f*)Bp0; b01 = *(const v2f*)(Bp0 + b4); }
      else       { b00.x = Bp0[0]; b00.y = Bp0[sBk]; b01.x = Bp0[b4]; b01.y = Bp0[b4 + sBk]; }
      if (SCALE) { a0.x *= as; a0.y *= as; a1.x *= as; a1.y *= as; }
      acc00 = __builtin_amdgcn_wmma_f32_16x16x4_f32(false, a0, false, b00,
                                                    (short)0, acc00, false, false);
      acc01 = __builtin_amdgcn_wmma_f32_16x16x4_f32(false, a1, false, b01,
                                                    (short)0, acc01, false, false);
      if (N2) {
        if (BCONT) { b10 = *(const v2f*)Bp1; b11 = *(const v2f*)(Bp1 + b4); }
        else       { b10.x = Bp1[0]; b10.y = Bp1[sBk]; b11.x = Bp1[b4]; b11.y = Bp1[b4 + sBk]; }
        acc10 = __builtin_amdgcn_wmma_f32_16x16x4_f32(false, a0, false, b10,
                                                      (short)0, acc10, false, false);
        acc11 = __builtin_amdgcn_wmma_f32_16x16x4_f32(false, a1, false, b11,
                                                      (short)0, acc11, false, false);
        Bp1 += 2 * b4;
      }
      Ap  += 2 * a4;
      Bp0 += 2 * b4;
    }

    float* Cp = Cb + (long long)cmBase * sCm + (long long)cn * sCn;
    const float* Rp =
        res ? res + (long long)tb * bsC + (long long)cmBase * sCm + (long long)cn * sCn
            : nullptr;
#pragma unroll
    for (int v = 0; v < 8; ++v) {
      if (cmBase + v < M) {
        float val0 = acc00[v] + acc01[v];
        if (bias) val0 += bias[cn];
        if (act == 1) {  // tanh-approx GELU (jax.nn.gelu default)
          float u = val0;
          float c = 0.7978845608028654f * (u + 0.044715f * u * u * u);
          val0 = 0.5f * u * (1.0f + tanhf(c));
        }
        if (Rp) val0 += Rp[0];
        Cp[0] = val0;
        if (N2) {
          float val1 = acc10[v] + acc11[v];
          if (bias) val1 += bias[cn + 16];
          if (act == 1) {
            float u = val1;
            float c = 0.7978845608028654f * (u + 0.044715f * u * u * u);
            val1 = 0.5f * u * (1.0f + tanhf(c));
          }
          if (Rp) val1 += Rp[16 * sCn];
          Cp[16 * sCn] = val1;
        }
      }
      Cp += sCm;
      if (Rp) Rp += sCm;
    }
  }
}

static void gemm(hipStream_t s, const float* A, const float* B, float* C,
                 const float* bias, const float* res,
                 int M, int N, int K,
                 long long sAm, long long sAk, long long bsA,
                 long long sBk, long long sBn, long long bsB,
                 long long sCm, long long sCn, long long bsC,
                 int batch, float alpha, int beta, int act)
{
  int tilesM = (M + 15) / 16, tilesN = N / 16;
  bool n2 = (tilesN % 2 == 0);
  bool sc = (alpha != 1.0f) || (M % 16 != 0);
  bool ac = (sAk == 1), bc = (sBk == 1);
  int tilesNe = n2 ? tilesN / 2 : tilesN;
  long long total = (long long)batch * tilesM * tilesNe;
  int blocks = (int)((total + 7) / 8);   // 8 waves / 256-thread block
#define FNO_ARGS A, B, C, bias, res, M, N, K, sAm, sAk, bsA, sBk, sBn, bsB, \
                 sCm, sCn, bsC, batch, alpha, beta, act, tilesM, tilesN
#define FNO_D(AC, BC, SC, NN) fno_wmma_gemm<AC, BC, SC, NN><<<blocks, 256, 0, s>>>(FNO_ARGS)
  if (n2) {
    if (sc) { if (ac) { bc ? FNO_D(true,true,true,true)   : FNO_D(true,false,true,true); }
              else    { bc ? FNO_D(false,true,true,true)  : FNO_D(false,false,true,true); } }
    else    { if (ac) { bc ? FNO_D(true,true,false,true)  : FNO_D(true,false,false,true); }
              else    { bc ? FNO_D(false,true,false,true) : FNO_D(false,false,false,true); } }
  } else {
    if (sc) { if (ac) { bc ? FNO_D(true,true,true,false)   : FNO_D(true,false,true,false); }
              else    { bc ? FNO_D(false,true,true,false)  : FNO_D(false,false,true,false); } }
    else    { if (ac) { bc ? FNO_D(true,true,false,false)  : FNO_D(true,false,false,false); }
              else    { bc ? FNO_D(false,true,false,false) : FNO_D(false,false,false,false); } }
  }
#undef FNO_D
#undef FNO_ARGS
}

extern "C" void kernel_launch(void* const* d_in, const int* in_sizes, int n_in,
                              void* d_out, int out_size, void* d_ws, size_t ws_size,
                              hipStream_t stream) {
  (void)in_sizes; (void)n_in; (void)out_size; (void)ws_size;
  const float* x      = (const float*)d_in[0];
  const float* w1r    = (const float*)d_in[1];
  const float* w1i    = (const float*)d_in[2];
  const float* w2r    = (const float*)d_in[3];
  const float* w2i    = (const float*)d_in[4];
  const float* skip_w = (const float*)d_in[5];
  const float* skip_b = (const float*)d_in[6];
  const float* mlp_w1 = (const float*)d_in[7];
  const float* mlp_b1 = (const float*)d_in[8];
  const float* mlp_w2 = (const float*)d_in[9];
  const float* mlp_b2 = (const float*)d_in[10];
  float* out = (float*)d_out;

  // Workspace layout (floats): ~88.3 MB
  float* ws   = (float*)d_ws;
  float* Fw   = ws;               // 8192
  float* Tre  = Fw   + 8192;      // 8192
  float* Tim  = Tre  + 8192;      // 8192
  float* iTre = Tim  + 8192;      // 8192
  float* iTim = iTre + 8192;      // 8192
  float* G    = iTim + 8192;      // 8192
  float* X3re = G    + 8192;      // [mode(512)][b(8)][c(64)] = 262144
  float* X3im = X3re + 262144;
  float* Y3re = X3im + 262144;    // [mode(512)][b(8)][o(64)]
  float* Y3im = Y3re + 262144;
  float* bigA = Y3im + 262144;    // 4,194,304 : Xw [bch(131072)][32], then Z [boh][32]
  float* hbuf = bigA + 4194304;   // 16,777,216 : [b(8)][hid(32)][hw(65536)]

  const long long HW = 65536, CHW = 64 * HW, HIDHW = 32 * HW;

  fno_fill_tables<<<128, 256, 0, stream>>>(Fw, Tre, Tim, iTre, iTim, G);

  // ---- Stage 1: DFT along W.  Xw[bch][0..15 re | 16..31 im] = x @ Fw
  gemm(stream, x, Fw, bigA, nullptr, nullptr, 131072, 32, 256,
       256, 1, 0,   32, 1, 0,   32, 1, 0,   1, 1.f, 0, 0);

  // ---- Stage 2: DFT along H (32 kept kx).  batch = 512 over (b,c)
  // X3 = T * Xw  (complex): C(m=kx,n=ky) -> X3[(kx*16+ky)*512 + b*64 + c]
  gemm(stream, Tre, bigA + 0,  X3re, nullptr, nullptr, 32, 16, 256,
       256, 1, 0,   32, 1, 8192,   8192, 512, 1,   512,  1.f, 0, 0);
  gemm(stream, Tim, bigA + 16, X3re, nullptr, nullptr, 32, 16, 256,
       256, 1, 0,   32, 1, 8192,   8192, 512, 1,   512, -1.f, 1, 0);
  gemm(stream, Tre, bigA + 16, X3im, nullptr, nullptr, 32, 16, 256,
       256, 1, 0,   32, 1, 8192,   8192, 512, 1,   512,  1.f, 0, 0);
  gemm(stream, Tim, bigA + 0,  X3im, nullptr, nullptr, 32, 16, 256,
       256, 1, 0,   32, 1, 8192,   8192, 512, 1,   512,  1.f, 1, 0);

  // ---- Stage 3: per-mode complex channel mix.  Y[b,o] = X[b,i] @ W[i,o]
  // corner 0 (kx 0..15) uses w1, corner 1 (kx 240..255) uses w2.  batch=256 each.
  // B strides over w*[i][o][kx][ky]: sBk(i)=16384, sBn(o)=256, bsB(mode)=1.
  for (int corner = 0; corner < 2; ++corner) {
    long long off = (long long)corner * 131072;  // 256 modes * 512
    const float* Wr = corner ? w2r : w1r;
    const float* Wi = corner ? w2i : w1i;
    gemm(stream, X3re + off, Wr, Y3re + off, nullptr, nullptr, 8, 64, 64,
         64, 1, 512,   16384, 256, 1,   64, 1, 512,   256,  1.f, 0, 0);
    gemm(stream, X3im + off, Wi, Y3re + off, nullptr, nullptr, 8, 64, 64,
         64, 1, 512,   16384, 256, 1,   64, 1, 512,   256, -1.f, 1, 0);
    gemm(stream, X3re + off, Wi, Y3im + off, nullptr, nullptr, 8, 64, 64,
         64, 1, 512,   16384, 256, 1,   64, 1, 512,   256,  1.f, 0, 0);
    gemm(stream, X3im + off, Wr, Y3im + off, nullptr, nullptr, 8, 64, 64,
         64, 1, 512,   16384, 256, 1,   64, 1, 512,   256,  1.f, 1, 0);
  }

  // ---- Stage 4: inverse DFT along H.  Z[boh][0..15 re|16..31 im] = iT * Y
  // batch = 512 over (b,o); reuses bigA (Xw is dead).
  gemm(stream, iTre, Y3re, bigA + 0,  nullptr, nullptr, 256, 16, 32,
       32, 1, 0,   8192, 512, 1,   32, 1, 8192,   512,  1.f, 0, 0);
  gemm(stream, iTim, Y3im, bigA + 0,  nullptr, nullptr, 256, 16, 32,
       32, 1, 0,   8192, 512, 1,   32, 1, 8192,   512, -1.f, 1, 0);
  gemm(stream, iTre, Y3im, bigA + 16, nullptr, nullptr, 256, 16, 32,
       32, 1, 0,   8192, 512, 1,   32, 1, 8192,   512,  1.f, 0, 0);
  gemm(stream, iTim, Y3re, bigA + 16, nullptr, nullptr, 256, 16, 32,
       32, 1, 0,   8192, 512, 1,   32, 1, 8192,   512,  1.f, 1, 0);

  // ---- Stage 5: real irfft synthesis along W.  spec = Z @ G  -> d_out
  gemm(stream, bigA, G, out, nullptr, nullptr, 131072, 256, 32,
       32, 1, 0,   256, 1, 0,   256, 1, 0,   1, 1.f, 0, 0);

  // ---- Stage 6a: x1 = spec + skip_w @ x + skip_b   (accumulate into d_out)
  gemm(stream, x, skip_w, out, skip_b, nullptr, 65536, 64, 64,
       1, HW, CHW,   1, 64, 0,   1, HW, CHW,   8, 1.f, 1, 0);

  // ---- Stage 6b: h = gelu(x1 @ mlp_w1^T + b1)  -> hbuf
  gemm(stream, out, mlp_w1, hbuf, mlp_b1, nullptr, 65536, 32, 64,
       1, HW, CHW,   1, 64, 0,   1, HW, HIDHW,   8, 1.f, 0, 1);

  // ---- Stage 6c: out = x + h @ mlp_w2^T + b2   (overwrites x1 in d_out)
  gemm(stream, hbuf, mlp_w2, out, mlp_b2, x, 65536, 64, 32,
       1, HW, HIDHW,   1, 32, 0,   1, HW, CHW,   8, 1.f, 0, 0);
}